// RF_scale_47888885350508
// MI455X (gfx1250) — compile-verified
//
#include <hip/hip_runtime.h>
#include <stdint.h>

// Problem constants (from reference): x is (8, 64, 128, 128) f32,
// out is (8, 64, 384, 384) f32. Fixed 3x-upsample stencil with reflect pad.
#define B_  8
#define C_  64
#define H_  128
#define W_  128
#define TILE_H 32
#define TILES_PER_PLANE (H_ / TILE_H)   // 4
#define OW  (3 * W_)                    // 384
#define LDS_ROWS (TILE_H + 2)           // 34

typedef __attribute__((ext_vector_type(4))) unsigned int u32x4;
typedef __attribute__((ext_vector_type(8))) int          i32x8;
typedef __attribute__((ext_vector_type(4))) int          i32x4;
typedef __attribute__((ext_vector_type(4))) float        f32x4;

__device__ __forceinline__ int reflect_h(int r) {
    r = (r < 0) ? -r : r;
    r = (r > H_ - 1) ? (2 * (H_ - 1) - r) : r;
    return r;
}
__device__ __forceinline__ int reflect_w(int c) {
    c = (c < 0) ? -c : c;
    c = (c > W_ - 1) ? (2 * (W_ - 1) - c) : c;
    return c;
}

__global__ __launch_bounds__(256, 4)
void rf_upsample3x_kernel(const float* __restrict__ x, float* __restrict__ out) {
    __shared__ float lds[LDS_ROWS * W_];

    const int tile  = blockIdx.x;
    const int plane = tile / TILES_PER_PLANE;      // b*C + c
    const int t     = tile % TILES_PER_PLANE;
    const int r0    = t * TILE_H;
    const int lo    = (r0 - 1 > 0) ? (r0 - 1) : 0;
    const int hi    = (r0 + TILE_H < H_ - 1) ? (r0 + TILE_H) : (H_ - 1);
    const int nrows = hi - lo + 1;                 // 33 or 34

    const float* src = x + (size_t)plane * (H_ * W_) + (size_t)lo * W_;

    // ---- Tensor Data Mover: DMA (nrows x 128) f32 tile -> LDS ------------
    // D# group0: count=1 | lds_addr | global_addr | type=2 ("image")
    // D# group1: data_size=4B, tensor_dim0=128, tensor_dim1=nrows,
    //            tile_dim0=128, tile_dim1=nrows, dim0_stride=128
    if (threadIdx.x < 32) {                        // one wave issues the DMA
        const uint64_t gaddr = (uint64_t)(uintptr_t)src;
        const uint32_t laddr = (uint32_t)(uintptr_t)lds;
        u32x4 g0;
        g0.x = 1u;                                           // count=1, user D#
        g0.y = laddr;                                        // lds_addr [63:32]
        g0.z = (uint32_t)(gaddr & 0xFFFFFFFFu);              // global_addr lo
        g0.w = (uint32_t)((gaddr >> 32) & 0x01FFFFFFu)       // global_addr hi
             | (2u << 30);                                   // type=2
        i32x8 g1;
        g1[0] = (int)(2u << 16);                   // data_size=2 (4B), mask=0
        g1[1] = (int)((uint32_t)W_ << 16);         // tensor_dim0[15:0] @bit48
        g1[2] = (int)((uint32_t)nrows << 16);      // tensor_dim1[15:0] @bit80
        g1[3] = (int)((uint32_t)W_ << 16);         // tile_dim0 @ [127:112]
        g1[4] = nrows;                             // tile_dim1 @ [143:128]
        g1[5] = W_;                                // tensor_dim0_stride lo32
        g1[6] = 0;                                 // stride hi / dim1_stride
        g1[7] = 0;
        const i32x4 gz4 = {0, 0, 0, 0};            // groups 2/3: unused dims
        const i32x8 gz8 = {0, 0, 0, 0, 0, 0, 0, 0};
#if __has_builtin(__builtin_amdgcn_tensor_load_to_lds)
        // amdgpu-toolchain (clang-23) 6-arg form:
        // (g0:u32x4, g1:i32x8, g2:i32x4, g3:i32x4, extra:i32x8, cpol:i32)
        __builtin_amdgcn_tensor_load_to_lds(g0, g1, gz4, gz4, gz8, 0);
#else
        asm volatile("tensor_load_to_lds %0, %1, %2, %3"
                     :: "s"(g0), "s"(g1), "s"(gz4), "s"(gz4) : "memory");
#endif
#if __has_builtin(__builtin_amdgcn_s_wait_tensorcnt)
        __builtin_amdgcn_s_wait_tensorcnt(0);
#else
        asm volatile("s_wait_tensorcnt 0x0" ::: "memory");
#endif
    }
    __syncthreads();

    // ---- Compute: 96 output rows per tile, one wave per row-job ----------
    float* outp = out + (size_t)plane * (3 * H_) * OW;
    const int wave = threadIdx.x >> 5;
    const int lane = threadIdx.x & 31;

    for (int j = wave; j < 3 * TILE_H; j += 8) {
        const int r  = r0 + j / 3;
        const int k1 = j % 3;                      // row sub-tap
        int prA, prB; float wxA, wxB;
        if (k1 == 0)      { prA = r - 1; prB = r;     wxA = 0.5f; wxB = 0.5f; }
        else if (k1 == 1) { prA = r;     prB = r;     wxA = 1.0f; wxB = 0.0f; }
        else              { prA = r;     prB = r + 1; wxA = 0.5f; wxB = 0.5f; }
        const float* rowA = &lds[(reflect_h(prA) - lo) * W_];
        const float* rowB = &lds[(reflect_h(prB) - lo) * W_];

        const int cw = lane * 4;                   // 4 input cols per lane
        float tt[6];
#pragma unroll
        for (int q = 0; q < 6; ++q) {              // cols cw-1 .. cw+4
            const int sc = reflect_w(cw - 1 + q);
            tt[q] = wxA * rowA[sc] + wxB * rowB[sc];
        }
        float o[12];
#pragma unroll
        for (int q = 0; q < 4; ++q) {              // 3 sub-cols per input col
            o[3 * q + 0] = 0.5f * (tt[q]     + tt[q + 1]);
            o[3 * q + 1] =         tt[q + 1];
            o[3 * q + 2] = 0.5f * (tt[q + 1] + tt[q + 2]);
        }
        // Output is written once and never re-read: stream past L2 (TH_NT).
        f32x4* orow = (f32x4*)(outp + (size_t)(3 * r + k1) * OW + cw * 3);
        __builtin_nontemporal_store((f32x4){o[0], o[1],  o[2],  o[3]}, orow + 0);
        __builtin_nontemporal_store((f32x4){o[4], o[5],  o[6],  o[7]}, orow + 1);
        __builtin_nontemporal_store((f32x4){o[8], o[9],  o[10], o[11]}, orow + 2);
    }
}

extern "C" void kernel_launch(void* const* d_in, const int* in_sizes, int n_in,
                              void* d_out, int out_size, void* d_ws, size_t ws_size,
                              hipStream_t stream) {
    const float* x   = (const float*)d_in[0];
    float*       out = (float*)d_out;
    dim3 grid(B_ * C_ * TILES_PER_PLANE);          // 2048 blocks
    rf_upsample3x_kernel<<<grid, 256, 0, stream>>>(x, out);
}